// WeightedAggregator_89489938580181
// MI455X (gfx1250) — compile-verified
//
#include <hip/hip_runtime.h>

// -------- CDNA5 (gfx1250, wave32) WeightedAggregator --------
// out[b,f] = sum_k w[b,k] * emb[samp[b,k], f],  w = row-normalized membership counts.
// One wave32 per batch row. Memory-bound: ~124 MB total traffic -> ~5.3us @ 23.3 TB/s.
// Final einsum done with V_WMMA_F32_16X16X4_F32 (broadcast-weight A rows; every D row
// equals the output f-chunk). 24 wmma per row; compute cost is negligible vs HBM.

typedef __attribute__((ext_vector_type(2))) float v2f;
typedef __attribute__((ext_vector_type(8))) float v8f;

#define K_S    11     // sampled neighborhood size (slot 0 = self)
#define PAD_D  32     // padded adjacency width
#define F_DIM  128    // feature dim
#define WPB    8      // waves (batch rows) per 256-thread block

__global__ __launch_bounds__(256)
void WeightedAggregator_kernel(const float* __restrict__ emb,
                               const int*   __restrict__ samp,
                               const int*   __restrict__ adj,
                               float*       __restrict__ out,
                               int batch)
{
    const int lane = threadIdx.x & 31;
    const int wave = __builtin_amdgcn_readfirstlane((int)(threadIdx.x >> 5));
    const int b    = blockIdx.x * WPB + wave;
    if (b >= batch) return;

    // ---- sampled node ids: wave-uniform addresses -> scalar loads ----
    const int* sp = samp + (size_t)b * K_S;
    int s[12];
#pragma unroll
    for (int v = 0; v < K_S; ++v) s[v] = sp[v];
    s[11] = s[0];                       // K pad (weight forced to 0)

    // ---- membership counts: lane d scans adjacency slot d (128B coalesced) ----
    int counts[K_S];
#pragma unroll
    for (int v = 0; v < K_S; ++v) counts[v] = 0;
#pragma unroll
    for (int n = 0; n < K_S; ++n) {
        const int a = adj[(size_t)s[n] * PAD_D + lane];
#pragma unroll
        for (int v = 0; v < K_S; ++v) {
            counts[v] += (__ballot(a == s[v]) != 0ull) ? 1 : 0;
        }
    }

    // ---- weights (BETA = 1): w = counts / sum(counts); sum >= 11 (self-loops) ----
    float w[12];
    float sum = 0.f;
#pragma unroll
    for (int v = 0; v < K_S; ++v) { w[v] = (float)counts[v]; sum += w[v]; }
    const float inv = 1.0f / sum;
#pragma unroll
    for (int v = 0; v < K_S; ++v) w[v] *= inv;
    w[11] = 0.f;

    // ---- aggregation via V_WMMA_F32_16X16X4_F32 ----
    // A[m,k]=w[k] (all rows equal), B[k,n]=emb[s[k], f0+n]  =>  D[m,n]=out[b,f0+n] for all m.
    // Source layout (ISA 7.12.2): lanes 0-15 hold K={k0,k0+1} in the 2 src VGPRs,
    // lanes 16-31 hold K={k0+2,k0+3}; n = lane & 15.
    const int  n  = lane & 15;
    const bool hi = (lane >> 4) != 0;
    float* op = out + (size_t)b * F_DIM;

#pragma unroll
    for (int fc = 0; fc < F_DIM / 16; ++fc) {
        const int f0 = fc * 16;
        v8f acc = {0.f, 0.f, 0.f, 0.f, 0.f, 0.f, 0.f, 0.f};
#pragma unroll
        for (int k0 = 0; k0 < 12; k0 += 4) {
            const int   s0 = hi ? s[k0 + 2] : s[k0];
            const int   s1 = hi ? s[k0 + 3] : s[k0 + 1];
            v2f A, B;
            A.x = hi ? w[k0 + 2] : w[k0];
            A.y = hi ? w[k0 + 3] : w[k0 + 1];
            B.x = emb[(size_t)s0 * F_DIM + f0 + n];   // 64B-contiguous per half-wave
            B.y = emb[(size_t)s1 * F_DIM + f0 + n];
            acc = __builtin_amdgcn_wmma_f32_16x16x4_f32(
                      false, A, false, B, (short)0, acc, false, false);
        }
        if (lane < 16) op[f0 + n] = acc[0];          // D row M=0 = output chunk
    }
}

extern "C" void kernel_launch(void* const* d_in, const int* in_sizes, int n_in,
                              void* d_out, int out_size, void* d_ws, size_t ws_size,
                              hipStream_t stream) {
    const float* emb  = (const float*)d_in[0];   // [N, 128] f32
    const int*   samp = (const int*)  d_in[1];   // [B, 11]  i32
    const int*   adj  = (const int*)  d_in[2];   // [N, 32]  i32
    float*       out  = (float*)d_out;           // [B, 128] f32

    const int batch  = in_sizes[1] / K_S;
    const int blocks = (batch + WPB - 1) / WPB;
    WeightedAggregator_kernel<<<blocks, 256, 0, stream>>>(emb, samp, adj, out, batch);
}